// EmRouting2d_83983790506264
// MI455X (gfx1250) — compile-verified
//
#include <hip/hip_runtime.h>
#include <math.h>

// EM-routing capsule forward, fully fused for MI455X (gfx1250).
// Strategy: never materialize v (462MB). One workgroup = 4 output positions.
// v 16x16 tiles are recomputed on demand with V_WMMA_F32_16X16X4_F32
// (M=16 rows = 4 positions x 4 pose-rows, N=16 = (c,m) tile, K=4 = j).
// W (0.6MB) stays L2-resident (192MB L2). LDS (~236KB of the 320KB WGP LDS)
// holds pose patches, routing weights r, sigma and small per-(pos,c) scalars.

typedef __attribute__((ext_vector_type(2))) float v2f;
typedef __attribute__((ext_vector_type(8))) float v8f;

namespace {
constexpr int BB   = 32;    // output caps
constexpr int PS   = 16;    // pose size
constexpr int KKA  = 288;   // 9*32
constexpr int NPOS = 4;     // positions per block (fills WMMA M=16)
constexpr int NTH  = 256;   // 8 waves (wave32), one c-tile per wave
constexpr float EPSF  = 1e-12f;
constexpr float LN2PI = 1.8378770664093453f;

// LDS layout (floats)
constexpr int P_OFF   = 0;                        // [4][288][16] pose patches
constexpr int A_OFF   = P_OFF   + NPOS*KKA*PS;    // [4][288]     a_in patches
constexpr int R_OFF   = A_OFF   + NPOS*KKA;       // [4][288][32] r / rp / ln_ap
constexpr int SIG_OFF = R_OFF   + NPOS*KKA*BB;    // [4][32][16]  sigma
constexpr int RS_OFF  = SIG_OFF + NPOS*BB*PS;     // [4][32]      r_sum
constexpr int IRS_OFF = RS_OFF  + NPOS*BB;        // [4][32]      1/(r_sum+eps)
constexpr int T_OFF   = IRS_OFF + NPOS*BB;        // [4][32]      -0.5*sum log(sig*ln2pi)
constexpr int LA_OFF  = T_OFF   + NPOS*BB;        // [4][32]      log a_out
constexpr int LDS_FLOATS = LA_OFF + NPOS*BB;      // 59008 floats = 236032 B <= 320KB
}

// Recompute one 16x16 v tile: D = A(16x4 f32) * B(4x16 f32), K=4.
// A from LDS pose patches, B gathered from W in L2.
__device__ __forceinline__ v8f pose_tile(const float* __restrict__ lp,
                                         const float* __restrict__ Wg,
                                         int a_off, int w_off) {
  v2f Af = *(const v2f*)(lp + a_off);     // lane: row=lane&15, K = {0,1} or {2,3}
  v2f Bf;
  Bf.x = Wg[w_off];                       // K = hf,   col n
  Bf.y = Wg[w_off + 8];                   // K = hf+2, col n
  v8f c = {};
  return __builtin_amdgcn_wmma_f32_16x16x4_f32(false, Af, false, Bf,
                                               (short)0, c, false, false);
}

extern "C" __global__ void __launch_bounds__(NTH)
em_routing_fused(const float* __restrict__ a_in,   // (16,32,14,14)
                 const float* __restrict__ pose,   // (16,512,14,14)
                 const float* __restrict__ Wg,     // (288,32,4,4)
                 const float* __restrict__ beta_u, // (32)
                 const float* __restrict__ beta_a, // (32)
                 float* __restrict__ out)          // [25088 a_map | 401408 pose_out]
{
  extern __shared__ float lds[];
  const int t    = threadIdx.x;
  const int lane = t & 31;
  const int wv   = t >> 5;        // wave id = c-tile (0..7)
  const int hf   = lane >> 4;     // lane half
  const int n    = lane & 15;     // column within tile
  const int m    = n & 3;         // pose column m
  const int c    = wv * 4 + (n >> 2);   // this lane's output capsule
  const int pos0 = hf * 2;        // C-tile rows: VGPR g(0..3) -> pos0, g(4..7) -> pos1
  const int pos1 = hf * 2 + 1;
  const int posA = n >> 2;        // A rows: row = lane&15 = pos*4+i
  const int iA   = n & 3;
  const int pbase = blockIdx.x * NPOS;  // 196 blocks * 4 positions = 16*49

  // ---- stage 0: gather unfolded patches into LDS --------------------------
  for (int e = t; e < NPOS*KKA*PS; e += NTH) {
    int ps = e & 15;
    int k  = (e >> 4) % KKA;
    int q  = e / (KKA*PS);
    int P  = pbase + q;
    int b  = P / 49, l = P % 49, oy = l / 7, ox = l % 7;
    int a  = k & 31, kk = k >> 5;
    int y  = oy*2 - 1 + kk/3, x = ox*2 - 1 + kk%3;
    float v = 0.f;
    if ((unsigned)y < 14u && (unsigned)x < 14u)
      v = pose[((b*512 + a*16 + ps)*14 + y)*14 + x];
    lds[P_OFF + e] = v;
  }
  for (int e = t; e < NPOS*KKA; e += NTH) {
    int k = e % KKA, q = e / KKA;
    int P = pbase + q;
    int b = P / 49, l = P % 49, oy = l / 7, ox = l % 7;
    int a = k & 31, kk = k >> 5;
    int y = oy*2 - 1 + kk/3, x = ox*2 - 1 + kk%3;
    float v = 0.f;
    if ((unsigned)y < 14u && (unsigned)x < 14u)
      v = a_in[((b*32 + a)*14 + y)*14 + x];
    lds[A_OFF + e] = v;
  }
  for (int e = t; e < NPOS*KKA*BB; e += NTH) lds[R_OFF + e] = 1.0f;
  __syncthreads();

  const int a_base = posA*KKA*PS + iA*4 + 2*hf;   // + k*16 per step
  const int w_base = c*16 + hf*4 + m;             // + k*512 per step
  const float lamv[3] = {0.01f*(1.f - 0.95f),
                         0.01f*(1.f - 0.95f*0.95f),
                         0.01f*(1.f - 0.95f*0.95f*0.95f)};
  float mu_r[8], i2s[8];

  for (int it = 0; it < 3; ++it) {
    const bool last = (it == 2);
    const float lam = lamv[it];

    // ---- m-step: rp = r * a_in ; r_sum over k ----------------------------
    for (int e = t; e < NPOS*KKA*BB; e += NTH)
      lds[R_OFF + e] *= lds[A_OFF + (e >> 5)];
    __syncthreads();
    if (t < NPOS*BB) {
      int pos = t >> 5;
      float s = 0.f;
      for (int k = 0; k < KKA; ++k)
        s += lds[R_OFF + (pos*KKA + k)*BB + (t & 31)];
      lds[RS_OFF + t]  = s;
      lds[IRS_OFF + t] = 1.f / (s + EPSF);
    }
    __syncthreads();

    // ---- m-step: one k-sweep, v recomputed by WMMA; S1=Σcv, S2=Σcv² ------
    float S1[8], S2[8];
    #pragma unroll
    for (int g = 0; g < 8; ++g) { S1[g] = 0.f; S2[g] = 0.f; }
    const float ir0 = lds[IRS_OFF + pos0*BB + c];
    const float ir1 = lds[IRS_OFF + pos1*BB + c];
    for (int k = 0; k < KKA; ++k) {
      v8f acc = pose_tile(lds + P_OFF, Wg, a_base + k*PS, w_base + k*512);
      float cf0 = lds[R_OFF + (pos0*KKA + k)*BB + c] * ir0;
      float cf1 = lds[R_OFF + (pos1*KKA + k)*BB + c] * ir1;
      #pragma unroll
      for (int g = 0; g < 4; ++g) { float v = acc[g]; S1[g] += cf0*v; S2[g] += cf0*v*v; }
      #pragma unroll
      for (int g = 4; g < 8; ++g) { float v = acc[g]; S1[g] += cf1*v; S2[g] += cf1*v*v; }
    }
    #pragma unroll
    for (int g = 0; g < 8; ++g) {
      int M = g + 8*hf, pos = M >> 2, ps = (M & 3)*4 + m;
      float mu = S1[g];
      float sg = S2[g] - mu*mu + EPSF;
      mu_r[g] = mu; i2s[g] = 0.5f / sg;
      lds[SIG_OFF + (pos*BB + c)*PS + ps] = sg;
      if (last) {
        int P = pbase + pos, b = P / 49, l = P % 49;
        out[25088 + ((b*512 + c*16 + ps)*7 + l/7)*7 + (l%7)] = mu;
      }
    }
    __syncthreads();

    // ---- m-step tail: activation a_out per (pos,c) -----------------------
    if (t < NPOS*BB) {
      int pos = t >> 5, cc = t & 31;
      float rs = lds[RS_OFF + t], sl = 0.f, cost = 0.f;
      for (int ps = 0; ps < PS; ++ps) {
        float sg = lds[SIG_OFF + (pos*BB + cc)*PS + ps];
        sl   += logf(sg * LN2PI);
        cost += beta_u[cc] + 0.5f*logf(sg);
      }
      cost *= rs;
      float ao = 1.f / (1.f + expf(-lam * (beta_a[cc] - cost)));
      lds[T_OFF + t]  = -0.5f * sl;
      lds[LA_OFF + t] = logf(ao);
      if (last) {
        int P = pbase + pos, b = P / 49, l = P % 49;
        out[((b*32 + cc)*7 + l/7)*7 + (l%7)] = ao;
      }
    }
    __syncthreads();
    if (last) break;

    // ---- e-step: second k-sweep, ln_ap into r buffer ---------------------
    const float TL0 = lds[T_OFF + pos0*BB + c] + lds[LA_OFF + pos0*BB + c];
    const float TL1 = lds[T_OFF + pos1*BB + c] + lds[LA_OFF + pos1*BB + c];
    for (int k = 0; k < KKA; ++k) {
      v8f acc = pose_tile(lds + P_OFF, Wg, a_base + k*PS, w_base + k*512);
      float sA = 0.f, sB = 0.f;
      #pragma unroll
      for (int g = 0; g < 4; ++g) { float d = acc[g] - mu_r[g]; sA += d*d*i2s[g]; }
      #pragma unroll
      for (int g = 4; g < 8; ++g) { float d = acc[g] - mu_r[g]; sB += d*d*i2s[g]; }
      sA += __shfl_xor(sA, 1, 32); sA += __shfl_xor(sA, 2, 32);  // sum over m
      sB += __shfl_xor(sB, 1, 32); sB += __shfl_xor(sB, 2, 32);
      if (m == 0) {
        lds[R_OFF + (pos0*KKA + k)*BB + c] = TL0 - sA;
        lds[R_OFF + (pos1*KKA + k)*BB + c] = TL1 - sB;
      }
    }
    __syncthreads();

    // ---- e-step: softmax over c per (pos,k) ------------------------------
    for (int row = t; row < NPOS*KKA; row += NTH) {
      int base = R_OFF + row*BB;
      float mx = -3.0e38f;
      for (int j = 0; j < BB; ++j) mx = fmaxf(mx, lds[base + j]);
      float s = 0.f;
      for (int j = 0; j < BB; ++j) { float e = expf(lds[base + j] - mx); lds[base + j] = e; s += e; }
      float inv = 1.f / s;
      for (int j = 0; j < BB; ++j) lds[base + j] *= inv;
    }
    __syncthreads();
  }
}

extern "C" void kernel_launch(void* const* d_in, const int* in_sizes, int n_in,
                              void* d_out, int out_size, void* d_ws, size_t ws_size,
                              hipStream_t stream) {
  (void)in_sizes; (void)n_in; (void)d_ws; (void)ws_size; (void)out_size;
  const float* a_in  = (const float*)d_in[0];
  const float* pose  = (const float*)d_in[1];
  const float* W     = (const float*)d_in[2];
  const float* bu    = (const float*)d_in[3];
  const float* ba    = (const float*)d_in[4];
  const size_t smem  = (size_t)LDS_FLOATS * sizeof(float);  // ~236KB of 320KB WGP LDS
  em_routing_fused<<<196, NTH, smem, stream>>>(a_in, pose, W, bu, ba, (float*)d_out);
}